// HarmonicGenerator_76811195122340
// MI455X (gfx1250) — compile-verified
//
#include <hip/hip_runtime.h>
#include <math.h>

#define SR        16000
#define NHARM     100
#define NFRAMES   500
#define NBATCH    4
#define NSAMP     80000                    // 500 * 16000/100
#define POS_SCALE (499.0 / 79999.0)        // linspace(0, 499, 80000) step
#define PHASE_C   (6.283185307179586476925286766559 / 16000.0) // 2*pi/SR
#define TILE      256
#define SCAN_T    1024

// ---- CDNA5 async-to-LDS path detection -------------------------------------
#if defined(__has_builtin)
#  if __has_builtin(__builtin_amdgcn_global_load_async_to_lds_b32)
#    define HAVE_ASYNC_LDS 1
#  endif
#  if __has_builtin(__builtin_amdgcn_s_wait_asynccnt)
#    define HAVE_WAIT_ASYNC 1
#  endif
#endif
#ifndef HAVE_ASYNC_LDS
#  define HAVE_ASYNC_LDS 0
#endif
#ifndef HAVE_WAIT_ASYNC
#  define HAVE_WAIT_ASYNC 0
#endif

typedef __attribute__((address_space(1))) int* gas1_i32;  // global (AS1)
typedef __attribute__((address_space(3))) int* las3_i32;  // LDS (AS3)

__device__ __forceinline__ void async_ld_f32(const float* g, float* l) {
#if HAVE_ASYNC_LDS
  // global_load_async_to_lds_b32: memory -> LDS, tracked by ASYNCcnt
  __builtin_amdgcn_global_load_async_to_lds_b32((gas1_i32)g, (las3_i32)l,
                                                /*offset=*/0, /*cpol=*/0);
#else
  *l = *g;  // sync fallback: global_load + ds_store
#endif
}

__device__ __forceinline__ void wait_async_lds() {
#if HAVE_WAIT_ASYNC
  __builtin_amdgcn_s_wait_asynccnt(0);
#elif HAVE_ASYNC_LDS
  asm volatile("s_wait_asynccnt 0" ::: "memory");
#endif
}

// ---- kernel: out[b,n] = sum_h amp_up * sin((h+1)*Phi + phi0) ---------------
// (placed first in the file so the disasm snippet shows the async staging)
__global__ __launch_bounds__(TILE)
void harm_kernel(const float* __restrict__ A,    // [B, H, F] amplitudes
                 const float* __restrict__ ph0,  // [B, H]    initial phase
                 const float* __restrict__ Phi,  // [B, N]
                 float* __restrict__ out) {      // [B, N]
  __shared__ float Alds[NHARM][4];  // 4 staged frame columns per tile
  __shared__ float P0[NHARM];

  const int b    = blockIdx.y;
  const int tile = blockIdx.x;
  const int n0   = tile * TILE;
  const int n    = n0 + threadIdx.x;

  const float* PhiB = Phi + b * NSAMP;
  // CDNA5 prefetch (global_prefetch_b8) for this thread's Phi element
  __builtin_prefetch(PhiB + min(n, NSAMP - 1), 0, 3);

  // Uniform base frame for the whole tile; a 256-sample tile spans
  // <= 1.60 frame positions, so i0 in [fbase, fbase+2], i1 <= fbase+3.
  const int fbase = (int)((double)n0 * POS_SCALE);

  const float* Ab = A + b * (NHARM * NFRAMES);
#pragma unroll
  for (int idx = threadIdx.x; idx < NHARM * 4; idx += TILE) {
    int h = idx >> 2, c = idx & 3;
    int col = min(fbase + c, NFRAMES - 1);
    async_ld_f32(Ab + h * NFRAMES + col, &Alds[h][c]);
  }
  if (threadIdx.x < NHARM)
    async_ld_f32(ph0 + b * NHARM + threadIdx.x, &P0[threadIdx.x]);

  wait_async_lds();
  __syncthreads();

  if (n >= NSAMP) return;

  const float phi = PhiB[n];
  float pos = (float)((double)n * POS_SCALE);
  int   i0  = (int)pos;                            // trunc == floor (pos >= 0)
  float w   = pos - (float)i0;
  const int c0 = i0 - fbase;                       // 0..2
  const int c1 = min(i0 + 1, NFRAMES - 1) - fbase; // <= 3

  float acc = 0.0f;
#pragma unroll 4
  for (int h = 0; h < NHARM; ++h) {
    float ang = fmaf((float)(h + 1), phi, P0[h]);
    float sv  = __sinf(ang);                 // v_sin_f32 (trans unit)
    float a0  = Alds[h][c0];                 // LDS broadcast across lanes
    float a1  = Alds[h][c1];
    float amp = fmaf(w, a1 - a0, a0);
    acc = fmaf(sv, amp, acc);
  }

  // nan_to_num(nan=0, posinf=0, neginf=0)
  if (!(fabsf(acc) < __builtin_inff())) acc = 0.0f;
  out[b * NSAMP + n] = acc;
}

// ---- per-sample phase increment given fp64 interp position -----------------
__device__ __forceinline__ float f0_inc_at(const float* __restrict__ f0s,
                                           double posd) {
  float pos = (float)posd;
  int   i0  = (int)pos;
  int   i1  = min(i0 + 1, NFRAMES - 1);
  float w   = pos - (float)i0;
  float f   = fmaf(w, f0s[i1] - f0s[i0], f0s[i0]);
  return (float)PHASE_C * f;
}

// ---- kernel: Phi[b,n] = inclusive cumsum over n of phase increments --------
__global__ __launch_bounds__(SCAN_T)
void phi_scan_kernel(const float* __restrict__ f0, float* __restrict__ Phi) {
  __shared__ float f0s[NFRAMES];
  __shared__ float part[SCAN_T];

  const int b = blockIdx.x;
  const int t = threadIdx.x;

  for (int i = t; i < NFRAMES; i += SCAN_T) f0s[i] = f0[b * NFRAMES + i];
  __syncthreads();

  const int chunk = (NSAMP + SCAN_T - 1) / SCAN_T;           // 79
  const int n0 = t * chunk;
  const int n1 = min(n0 + chunk, NSAMP);
  const double pos0 = (double)n0 * POS_SCALE;  // one f64 mul; loop uses adds

  float  s    = 0.0f;
  double posd = pos0;
  for (int n = n0; n < n1; ++n, posd += POS_SCALE)
    s += f0_inc_at(f0s, posd);
  part[t] = s;
  __syncthreads();

  // Hillis-Steele inclusive scan over 1024 partials
  for (int off = 1; off < SCAN_T; off <<= 1) {
    float v = (t >= off) ? part[t - off] : 0.0f;
    __syncthreads();
    part[t] += v;
    __syncthreads();
  }

  float run = part[t] - s;  // exclusive prefix for this thread's chunk
  float* Pb = Phi + b * NSAMP;
  posd = pos0;
  for (int n = n0; n < n1; ++n, posd += POS_SCALE) {
    run += f0_inc_at(f0s, posd);
    Pb[n] = run;            // inclusive cumsum, matches jnp.cumsum
  }
}

extern "C" void kernel_launch(void* const* d_in, const int* in_sizes, int n_in,
                              void* d_out, int out_size, void* d_ws, size_t ws_size,
                              hipStream_t stream) {
  const float* f0            = (const float*)d_in[0];  // [4, 500]
  const float* amplitudes    = (const float*)d_in[1];  // [4, 100, 500]
  const float* initial_phase = (const float*)d_in[2];  // [4, 100]
  float* out = (float*)d_out;                          // [4, 80000]
  float* Phi = (float*)d_ws;                           // [4, 80000] scratch

  phi_scan_kernel<<<dim3(NBATCH), dim3(SCAN_T), 0, stream>>>(f0, Phi);

  dim3 grid((NSAMP + TILE - 1) / TILE, NBATCH);
  harm_kernel<<<grid, dim3(TILE), 0, stream>>>(amplitudes, initial_phase, Phi, out);
}